// SpiralNet_60619168416159
// MI455X (gfx1250) — compile-verified
//
#include <hip/hip_runtime.h>
#include <hip/hip_bf16.h>
#include <math.h>

typedef __attribute__((ext_vector_type(2))) float v2f;
typedef __attribute__((ext_vector_type(8))) float v8f;

// ---------------------------------------------------------------------------
// Gather + GEMM: y[row, o] = sum_k feat[row, k] * W[o, k] + bias[o]
//   feat[row = b*V + v, k = s*CIN + c] = hin[b, spirals[b,v,s], c]
// One block = 8 wave32 = 256 threads handles a 16-row tile and all COUT cols.
// Uses V_WMMA_F32_16X16X4_F32 (fp32 tensor path, native precision).
// ---------------------------------------------------------------------------
template <int CIN, int COUT>
__launch_bounds__(256)
__global__ void spiral_gemm(const float* __restrict__ hin,
                            const int* __restrict__ spirals,
                            const float* __restrict__ W,
                            const float* __restrict__ bias,
                            float* __restrict__ y,
                            int V)
{
    constexpr int S  = 9;
    constexpr int K  = S * CIN;       // 27 / 576 / 1152
    constexpr int KC = 192;           // K-chunk staged in LDS (multiple of 4)
    constexpr int NT = COUT >> 4;     // # of 16-wide N tiles (4 / 8 / 16)

    __shared__ float lds[16 * KC];    // 12 KB feature tile (zero padded)

    const int rowTile  = blockIdx.x;
    const int row_base = rowTile * 16;
    const int b        = row_base / V;       // V % 16 == 0: tile stays in batch
    const int v_base   = row_base - b * V;

    const int*   sp = spirals + (size_t)b * V * S;
    const float* hb = hin     + (size_t)b * V * CIN;

    const int tid  = threadIdx.x;
    const int wid  = tid >> 5;        // wave id 0..7
    const int lane = tid & 31;
    const int half = lane >> 4;       // 0: K pair {0,1}; 1: K pair {2,3}
    const int lrow = lane & 15;       // M for A frag / N-in-tile for B frag

    const int nt0 = wid;              // first N tile for this wave
    const int nt1 = wid + 8;          // second N tile (COUT==256 only)
    v8f acc0 = {};
    v8f acc1 = {};

    for (int k0 = 0; k0 < K; k0 += KC) {
        __syncthreads();
        // ---- stage 16 x KC feature tile into LDS (zero pad past K) ----
        for (int e = tid; e < 16 * KC; e += 256) {
            const int r  = e / KC;
            const int kk = e - r * KC;
            const int k  = k0 + kk;
            float val = 0.0f;
            if (k < K) {
                const int s   = k / CIN;          // compile-time-const divisor
                const int c   = k - s * CIN;
                const int idx = sp[(v_base + r) * S + s];
                val = hb[(size_t)idx * CIN + c];
            }
            lds[r * KC + kk] = val;
        }
        __syncthreads();

        // ---- WMMA sweep over this K chunk ----
        for (int kk = 0; kk < KC && (k0 + kk) < K; kk += 4) {
            v2f a;
            a.x = lds[lrow * KC + kk + 2 * half];
            a.y = lds[lrow * KC + kk + 2 * half + 1];
            const int kg = k0 + kk + 2 * half;

            if (nt0 < NT) {                       // wave-uniform: EXEC stays full
                const float* wr = W + (size_t)((nt0 << 4) + lrow) * K;
                v2f bf;
                bf.x = (kg     < K) ? wr[kg]     : 0.0f;
                bf.y = (kg + 1 < K) ? wr[kg + 1] : 0.0f;
                acc0 = __builtin_amdgcn_wmma_f32_16x16x4_f32(
                    false, a, false, bf, (short)0, acc0, false, false);
            }
            if (nt1 < NT) {                       // wave-uniform
                const float* wr = W + (size_t)((nt1 << 4) + lrow) * K;
                v2f bf;
                bf.x = (kg     < K) ? wr[kg]     : 0.0f;
                bf.y = (kg + 1 < K) ? wr[kg + 1] : 0.0f;
                acc1 = __builtin_amdgcn_wmma_f32_16x16x4_f32(
                    false, a, false, bf, (short)0, acc1, false, false);
            }
        }
    }

    // ---- store D: lanes 0-15 rows 0-7, lanes 16-31 rows 8-15 ----
    if (nt0 < NT) {
        const int col = (nt0 << 4) + lrow;
        const float bv = bias[col];
#pragma unroll
        for (int r = 0; r < 8; ++r) {
            const int m = r + 8 * half;
            y[(size_t)(row_base + m) * COUT + col] = acc0[r] + bv;
        }
    }
    if (nt1 < NT) {
        const int col = (nt1 << 4) + lrow;
        const float bv = bias[col];
#pragma unroll
        for (int r = 0; r < 8; ++r) {
            const int m = r + 8 * half;
            y[(size_t)(row_base + m) * COUT + col] = acc1[r] + bv;
        }
    }
}

// ---------------------------------------------------------------------------
// BatchNorm stats per channel over N rows: fold into scale/shift (a, b).
// ---------------------------------------------------------------------------
__launch_bounds__(256)
__global__ void bn_stats(const float* __restrict__ y,
                         const float* __restrict__ g,
                         const float* __restrict__ bt,
                         float* __restrict__ ab,
                         int N, int Cout)
{
    const int ch = blockIdx.x;
    float s = 0.0f, s2 = 0.0f;
    for (int r = threadIdx.x; r < N; r += 256) {
        const float v = y[(size_t)r * Cout + ch];
        s  += v;
        s2 += v * v;
    }
    __shared__ float sh1[256];
    __shared__ float sh2[256];
    sh1[threadIdx.x] = s;
    sh2[threadIdx.x] = s2;
    __syncthreads();
    for (int off = 128; off > 0; off >>= 1) {
        if (threadIdx.x < off) {
            sh1[threadIdx.x] += sh1[threadIdx.x + off];
            sh2[threadIdx.x] += sh2[threadIdx.x + off];
        }
        __syncthreads();
    }
    if (threadIdx.x == 0) {
        const float inv_n = 1.0f / (float)N;
        const float mean  = sh1[0] * inv_n;
        const float var   = sh2[0] * inv_n - mean * mean;
        const float rstd  = rsqrtf(var + 1e-5f);
        const float a     = g[ch] * rstd;
        ab[ch]        = a;
        ab[Cout + ch] = bt[ch] - mean * a;
    }
}

__launch_bounds__(256)
__global__ void bn_apply_relu(float* __restrict__ y,
                              const float* __restrict__ ab,
                              int n, int Cout)
{
    const int i = blockIdx.x * 256 + threadIdx.x;
    if (i < n) {
        const int ch = i % Cout;
        const float v = y[i] * ab[ch] + ab[Cout + ch];
        y[i] = fmaxf(v, 0.0f);
    }
}

// ---------------------------------------------------------------------------
// Max pool over V (two stage). ReLU output => 0 is a valid identity element.
// ---------------------------------------------------------------------------
__launch_bounds__(256)
__global__ void maxpool_partial(const float* __restrict__ h,
                                float* __restrict__ part,
                                int V, int C, int chunks)
{
    const int b     = blockIdx.x;
    const int chunk = blockIdx.y;
    const int c     = threadIdx.x;            // C == 256 == blockDim.x
    const int rows  = (V + chunks - 1) / chunks;
    const int v0    = chunk * rows;
    const int v1    = min(v0 + rows, V);
    float m = 0.0f;
    const float* p = h + (size_t)b * V * C + c;
    for (int v = v0; v < v1; ++v)
        m = fmaxf(m, p[(size_t)v * C]);
    part[((size_t)b * chunks + chunk) * C + c] = m;
}

__launch_bounds__(256)
__global__ void maxpool_reduce(const float* __restrict__ part,
                               float* __restrict__ pooled,
                               int C, int chunks)
{
    const int b = blockIdx.x;
    const int c = threadIdx.x;
    float m = 0.0f;
    for (int k = 0; k < chunks; ++k)
        m = fmaxf(m, part[((size_t)b * chunks + k) * C + c]);
    pooled[b * C + c] = m;
}

// ---------------------------------------------------------------------------
// Final linear: out[b,o] = pooled[b,:] . pw[o,:] + pb[o]   (8x256x256, tiny)
// ---------------------------------------------------------------------------
__launch_bounds__(256)
__global__ void final_linear(const float* __restrict__ pooled,
                             const float* __restrict__ pw,
                             const float* __restrict__ pb,
                             float* __restrict__ out,
                             int Bn, int C)
{
    const int i = blockIdx.x * 256 + threadIdx.x;
    if (i < Bn * C) {
        const int b = i / C;
        const int o = i - b * C;
        const float* pr = pooled + (size_t)b * C;
        const float* wr = pw + (size_t)o * C;
        float s = pb[o];
        for (int k = 0; k < C; ++k)
            s += pr[k] * wr[k];
        out[i] = s;
    }
}

// ---------------------------------------------------------------------------
extern "C" void kernel_launch(void* const* d_in, const int* in_sizes, int n_in,
                              void* d_out, int out_size, void* d_ws, size_t ws_size,
                              hipStream_t stream)
{
    (void)in_sizes; (void)n_in; (void)out_size; (void)ws_size;

    const float* x       = (const float*)d_in[0];
    const int*   spirals = (const int*)  d_in[1];
    const float* w1  = (const float*)d_in[2];
    const float* b1  = (const float*)d_in[3];
    const float* g1  = (const float*)d_in[4];
    const float* bt1 = (const float*)d_in[5];
    const float* w2  = (const float*)d_in[6];
    const float* b2  = (const float*)d_in[7];
    const float* g2  = (const float*)d_in[8];
    const float* bt2 = (const float*)d_in[9];
    const float* w3  = (const float*)d_in[10];
    const float* b3  = (const float*)d_in[11];
    const float* g3  = (const float*)d_in[12];
    const float* bt3 = (const float*)d_in[13];
    const float* pw  = (const float*)d_in[14];
    const float* pb  = (const float*)d_in[15];
    float* out = (float*)d_out;

    const int B = 8, V = 12000;
    const int N = B * V;            // 96000 rows, divisible by 16
    const int CHUNKS = 25;

    // workspace layout (fp32), ~172 MB total
    float* h1     = (float*)d_ws;                 // N * 64
    float* h2     = h1 + (size_t)N * 64;          // N * 128
    float* h3     = h2 + (size_t)N * 128;         // N * 256
    float* ab     = h3 + (size_t)N * 256;         // 2 * 256
    float* pooled = ab + 512;                     // B * 256
    float* part   = pooled + (size_t)B * 256;     // B * CHUNKS * 256

    const dim3 blk(256);
    const int rowTiles = N / 16;                  // 6000

    // ---- layer 1: CIN=3, COUT=64, K=27 ----
    spiral_gemm<3, 64><<<rowTiles, blk, 0, stream>>>(x, spirals, w1, b1, h1, V);
    bn_stats<<<64, blk, 0, stream>>>(h1, g1, bt1, ab, N, 64);
    bn_apply_relu<<<(N * 64 + 255) / 256, blk, 0, stream>>>(h1, ab, N * 64, 64);

    // ---- layer 2: CIN=64, COUT=128, K=576 ----
    spiral_gemm<64, 128><<<rowTiles, blk, 0, stream>>>(h1, spirals, w2, b2, h2, V);
    bn_stats<<<128, blk, 0, stream>>>(h2, g2, bt2, ab, N, 128);
    bn_apply_relu<<<(N * 128 + 255) / 256, blk, 0, stream>>>(h2, ab, N * 128, 128);

    // ---- layer 3: CIN=128, COUT=256, K=1152 ----
    spiral_gemm<128, 256><<<rowTiles, blk, 0, stream>>>(h2, spirals, w3, b3, h3, V);
    bn_stats<<<256, blk, 0, stream>>>(h3, g3, bt3, ab, N, 256);
    bn_apply_relu<<<(N * 256 + 255) / 256, blk, 0, stream>>>(h3, ab, N * 256, 256);

    // ---- global max pool over V, then final linear ----
    maxpool_partial<<<dim3(B, CHUNKS), blk, 0, stream>>>(h3, part, V, 256, CHUNKS);
    maxpool_reduce<<<B, blk, 0, stream>>>(part, pooled, 256, CHUNKS);
    final_linear<<<(B * 256 + 255) / 256, blk, 0, stream>>>(pooled, pw, pb, out, B, 256);
}